// SAE_16088947491065
// MI455X (gfx1250) — compile-verified
//
#include <hip/hip_runtime.h>

// ---------------------------------------------------------------------------
// TopK-SAE forward for MI455X (gfx1250, wave32).
//   encode: h = relu((x - b_dec) @ W_enc + b_enc)   [4096 x 32768]  (f32 WMMA)
//           M-blocked 2x per wave: 4 B-loads feed 4 WMMAs (VMEM-issue relief)
//   top-64 per row (streaming, register-resident, wave-cooperative)
//   decode: out = b_dec + sum_k act_k * W_dec[id_k]  (sparse, memory-bound)
// ---------------------------------------------------------------------------

#define D_IN      2048
#define D_SAE     32768
#define KSEL      64
#define BATCH     4096
#define M_TILE    32            // batch rows per workgroup (2 WMMA row-tiles)
#define CHUNK     256           // feature columns per chunk (8 waves x 32 cols)
#define NCHUNK    (D_SAE / CHUNK)   // 128
#define KB        512           // K-block staged in LDS
#define XS_STRIDE (KB + 4)      // 516: lane m -> banks {4m,4m+1}, conflict-free b64
#define HB_STRIDE 260           // 260: lane-halves hit disjoint bank sets

typedef __attribute__((ext_vector_type(2))) float v2f;
typedef __attribute__((ext_vector_type(8))) float v8f;

__global__ __launch_bounds__(256) void sae_encode_topk(
    const float* __restrict__ x, const float* __restrict__ W_enc,
    const float* __restrict__ b_enc, const float* __restrict__ b_dec,
    float* __restrict__ top_vals, int* __restrict__ top_ids)
{
    __shared__ __align__(16) float xs[M_TILE * XS_STRIDE];   // (x - b_dec) K-block
    __shared__ __align__(16) float hbuf[M_TILE * HB_STRIDE]; // relu'd h chunk

    const int tid  = threadIdx.x;
    const int lane = tid & 31;
    const int w    = tid >> 5;          // wave id 0..7
    const int hi   = lane >> 4;         // lane half (WMMA layout)
    const int nlo  = lane & 15;
    const int row0 = blockIdx.x * M_TILE;

    const float* xsrow0 = &xs[nlo * XS_STRIDE + 2 * hi];          // rows 0..15
    const float* xsrow1 = &xs[(16 + nlo) * XS_STRIDE + 2 * hi];   // rows 16..31

    // Streaming top-64 state. Wave w owns batch rows 4w .. 4w+3.
    // 64 slots = 2 per lane (slot 'lane' and slot 'lane+32'), kept in VGPRs.
    float tv0[4] = {-1e30f, -1e30f, -1e30f, -1e30f};
    float tv1[4] = {-1e30f, -1e30f, -1e30f, -1e30f};
    int   ti0[4] = {0, 0, 0, 0}, ti1[4] = {0, 0, 0, 0};
    float rmin[4] = {-1e30f, -1e30f, -1e30f, -1e30f};

    for (int ch = 0; ch < NCHUNK; ++ch) {
        const int cbase = ch * CHUNK;
        const int n0    = w * 32;       // this wave's 32 columns within chunk

        // B frag pointer: W_enc[(k + 2*hi)*D_SAE + cbase + n0 + nlo]
        const float* pb = W_enc + (size_t)(2 * hi) * D_SAE + (cbase + n0 + nlo);

        v8f c00 = {}; v8f c01 = {};   // row-tile 0, col-tiles 0/1
        v8f c10 = {}; v8f c11 = {};   // row-tile 1, col-tiles 0/1

        for (int kb = 0; kb < D_IN; kb += KB) {
            __syncthreads();   // previous k-block readers (and prev-chunk scan) done
            // Cooperative stage: xs = x_tile[:, kb:kb+KB] - b_dec[kb:kb+KB]
            for (int e = tid * 4; e < M_TILE * KB; e += 256 * 4) {
                const int r = e / KB, c = e % KB;
                const float4 xv = *(const float4*)(x + (size_t)(row0 + r) * D_IN + kb + c);
                const float4 bd = *(const float4*)(b_dec + kb + c);
                float4 o; o.x = xv.x - bd.x; o.y = xv.y - bd.y;
                          o.z = xv.z - bd.z; o.w = xv.w - bd.w;
                *(float4*)(&xs[r * XS_STRIDE + c]) = o;
            }
            __syncthreads();

            #pragma unroll 4
            for (int k = 0; k < KB; k += 4) {
                // A frags (16x4 f32): ds_load_b64 per row-tile (pairs merge to 2addr)
                const v2f a0 = *(const v2f*)(xsrow0 + k);
                const v2f a1 = *(const v2f*)(xsrow1 + k);
                // B frags (4x16 f32): VGPR0 = K rows {k, k+2}, VGPR1 = {k+1, k+3};
                // shared by both row-tiles -> 4 global loads feed 4 WMMAs.
                v2f b0, b1;
                b0.x = pb[0];   b0.y = pb[D_SAE];
                b1.x = pb[16];  b1.y = pb[D_SAE + 16];
                c00 = __builtin_amdgcn_wmma_f32_16x16x4_f32(false, a0, false, b0,
                                                            (short)0, c00, false, false);
                c01 = __builtin_amdgcn_wmma_f32_16x16x4_f32(false, a0, false, b1,
                                                            (short)0, c01, false, false);
                c10 = __builtin_amdgcn_wmma_f32_16x16x4_f32(false, a1, false, b0,
                                                            (short)0, c10, false, false);
                c11 = __builtin_amdgcn_wmma_f32_16x16x4_f32(false, a1, false, b1,
                                                            (short)0, c11, false, false);
                pb += (size_t)4 * D_SAE;
            }
        }

        // bias + ReLU, stage h chunk through LDS for the row-major top-k scan
        const float be0 = b_enc[cbase + n0 + nlo];
        const float be1 = b_enc[cbase + n0 + 16 + nlo];
        __syncthreads();
        #pragma unroll
        for (int r = 0; r < 8; ++r) {
            const int row = r + 8 * hi;           // C layout: VGPR r -> M = r / r+8
            hbuf[row * HB_STRIDE + n0 + nlo]             = fmaxf(c00[r] + be0, 0.0f);
            hbuf[row * HB_STRIDE + n0 + 16 + nlo]        = fmaxf(c01[r] + be1, 0.0f);
            hbuf[(16 + row) * HB_STRIDE + n0 + nlo]      = fmaxf(c10[r] + be0, 0.0f);
            hbuf[(16 + row) * HB_STRIDE + n0 + 16 + nlo] = fmaxf(c11[r] + be1, 0.0f);
        }
        __syncthreads();

        // Top-k merge: wave w scans rows 4w .. 4w+3 across all 256 chunk columns.
        #pragma unroll
        for (int rr = 0; rr < 4; ++rr) {
            const int row = 4 * w + rr;
            for (int t = 0; t < 8; ++t) {
                float v  = hbuf[row * HB_STRIDE + t * 32 + lane];
                int  idx = cbase + t * 32 + lane;
                bool cand = v > rmin[rr];
                unsigned long long m;
                while ((m = __ballot(cand)) != 0ull) {
                    const int src = __ffsll(m) - 1;
                    const float cv = __shfl(v, src, 32);
                    const int   ci = __shfl(idx, src, 32);
                    // wave-wide argmin over the 64 register slots
                    float mv; int ms;
                    if (tv0[rr] <= tv1[rr]) { mv = tv0[rr]; ms = lane; }
                    else                    { mv = tv1[rr]; ms = lane + 32; }
                    #pragma unroll
                    for (int off = 16; off > 0; off >>= 1) {
                        const float ov = __shfl_xor(mv, off, 32);
                        const int   os = __shfl_xor(ms, off, 32);
                        if (ov < mv || (ov == mv && os < ms)) { mv = ov; ms = os; }
                    }
                    if (cv > mv) {
                        if (ms == lane)           { tv0[rr] = cv; ti0[rr] = ci; }
                        else if (ms == lane + 32) { tv1[rr] = cv; ti1[rr] = ci; }
                    }
                    // new running min -> wave-uniform threshold
                    float nm = fminf(tv0[rr], tv1[rr]);
                    #pragma unroll
                    for (int off = 16; off > 0; off >>= 1)
                        nm = fminf(nm, __shfl_xor(nm, off, 32));
                    rmin[rr] = nm;
                    if (lane == src) cand = false;
                    cand = cand && (v > rmin[rr]);
                }
            }
        }
    }

    // Emit per-row top-64 (unsorted; decode is order-invariant).
    #pragma unroll
    for (int rr = 0; rr < 4; ++rr) {
        const int grow = row0 + 4 * w + rr;
        top_vals[grow * KSEL + lane]      = tv0[rr];
        top_vals[grow * KSEL + lane + 32] = tv1[rr];
        top_ids [grow * KSEL + lane]      = ti0[rr];
        top_ids [grow * KSEL + lane + 32] = ti1[rr];
    }
}

// Sparse decode: out[b,:] = b_dec + sum_k val_k * W_dec[id_k, :]
__global__ __launch_bounds__(256) void sae_decode(
    const float* __restrict__ top_vals, const int* __restrict__ top_ids,
    const float* __restrict__ W_dec, const float* __restrict__ b_dec,
    float* __restrict__ out)
{
    __shared__ float sv[KSEL];
    __shared__ int   si[KSEL];
    const int b = blockIdx.x, tid = threadIdx.x;
    if (tid < KSEL) { sv[tid] = top_vals[b * KSEL + tid];
                      si[tid] = top_ids [b * KSEL + tid]; }
    __syncthreads();

    const int col = tid * 8;                       // 256 threads x 8 cols = 2048
    float4 acc0 = *(const float4*)(b_dec + col);
    float4 acc1 = *(const float4*)(b_dec + col + 4);

    #pragma unroll 4
    for (int k = 0; k < KSEL; ++k) {
        const float a = sv[k];
        const float* wr = W_dec + (size_t)si[k] * D_IN + col;
        const float4 w0 = *(const float4*)(wr);
        const float4 w1 = *(const float4*)(wr + 4);
        acc0.x = fmaf(a, w0.x, acc0.x); acc0.y = fmaf(a, w0.y, acc0.y);
        acc0.z = fmaf(a, w0.z, acc0.z); acc0.w = fmaf(a, w0.w, acc0.w);
        acc1.x = fmaf(a, w1.x, acc1.x); acc1.y = fmaf(a, w1.y, acc1.y);
        acc1.z = fmaf(a, w1.z, acc1.z); acc1.w = fmaf(a, w1.w, acc1.w);
    }
    float* op = out + (size_t)b * D_IN + col;
    *(float4*)(op)     = acc0;
    *(float4*)(op + 4) = acc1;
}

extern "C" void kernel_launch(void* const* d_in, const int* in_sizes, int n_in,
                              void* d_out, int out_size, void* d_ws, size_t ws_size,
                              hipStream_t stream) {
    const float* x     = (const float*)d_in[0];
    const float* W_enc = (const float*)d_in[1];
    const float* b_enc = (const float*)d_in[2];
    const float* W_dec = (const float*)d_in[3];
    const float* b_dec = (const float*)d_in[4];
    float* out = (float*)d_out;

    // workspace: top_vals [B*64] f32, then top_ids [B*64] i32  (2 MB total)
    float* top_vals = (float*)d_ws;
    int*   top_ids  = (int*)((char*)d_ws + (size_t)BATCH * KSEL * sizeof(float));

    sae_encode_topk<<<BATCH / M_TILE, 256, 0, stream>>>(x, W_enc, b_enc, b_dec,
                                                        top_vals, top_ids);
    sae_decode<<<BATCH, 256, 0, stream>>>(top_vals, top_ids, W_dec, b_dec, out);
}